// GaussianConv2d_56092272886226
// MI455X (gfx1250) — compile-verified
//
#include <hip/hip_runtime.h>
#include <hip/hip_bf16.h>
#include <stdint.h>

// Depthwise 7x7 Gaussian blur, fp32, zero padding, fused separable passes.
// Bandwidth-bound (~201 MB @ 23.3 TB/s ~= 8.6 us floor).
// Data movement uses CDNA5 paths:
//   - interior tiles: TDM tensor_load_to_lds (one DMA per workgroup, TENSORcnt)
//   - border tiles:   per-lane GLOBAL_LOAD_ASYNC_TO_LDS_B128 (ASYNCcnt) with
//                     LDS zero-prefill for the padding region.

#define KW      7
#define HALO    3
#define TILE    64
#define SROWS   (TILE + 2 * HALO)   // 70 halo rows
#define SCOLS   (TILE + 8)          // 72 cols: [x0-4, x0+68), keeps 16B chunks aligned
#define NCHUNK  (SCOLS / 4)         // 18 x b128 chunks per row
#define IMG_W   512
#define IMG_H   512

typedef uint32_t u32x4 __attribute__((ext_vector_type(4)));
typedef int      i32x8 __attribute__((ext_vector_type(8)));
typedef int      i32x4 __attribute__((ext_vector_type(4)));

__global__ __launch_bounds__(256)
void gauss7_tile_kernel(const float* __restrict__ x,
                        const float* __restrict__ sig,
                        float* __restrict__ out)
{
    __shared__ float S[SROWS * SCOLS];   // raw halo tile    (20160 B)
    __shared__ float T[SROWS * TILE];    // after horiz pass (17920 B)

    const int tid   = threadIdx.x;
    const int x0    = blockIdx.x * TILE;
    const int y0    = blockIdx.y * TILE;
    const int plane = blockIdx.z;

    const float* __restrict__ src = x   + (size_t)plane * (IMG_W * IMG_H);
    float*       __restrict__ dst = out + (size_t)plane * (IMG_W * IMG_H);

    // LDS byte offset of S (low 32 bits of the generic shared-aperture address)
    const uint32_t s_base = (uint32_t)(uintptr_t)(const void*)&S[0];

    // ---- 1D Gaussian weights from runtime sigma (2D kernel factors exactly) ----
    const float sigma  = sig[0];
    const float inv2s2 = -1.0f / (2.0f * sigma * sigma);
    float w[KW];
    float wsum = 0.0f;
    #pragma unroll
    for (int k = 0; k < KW; ++k) {
        float d = (float)(k - HALO);
        w[k] = __expf(d * d * inv2s2);
        wsum += w[k];
    }
    const float wn = 1.0f / wsum;
    #pragma unroll
    for (int k = 0; k < KW; ++k) w[k] *= wn;

    const bool border = (x0 == 0) | (y0 == 0) |
                        (x0 + TILE == IMG_W) | (y0 + TILE == IMG_H);

    if (border) {
        // ---- zero-prefill for the padding region (uniform branch) ----
        for (int i = tid; i < SROWS * SCOLS; i += 256) S[i] = 0.0f;
        __syncthreads();

        // ---- per-lane async global->LDS staging, OOB chunks skipped ----
        // Chunk grid is 4-float aligned in x; image borders (0, 512) are
        // 4-aligned, so each 16B chunk is entirely in-bounds or entirely pad.
        for (int i = tid; i < SROWS * NCHUNK; i += 256) {
            const int row = i / NCHUNK;
            const int cx  = i - row * NCHUNK;
            const int gy  = y0 - HALO + row;
            const int gx  = x0 - 4 + cx * 4;
            const bool valid = (gy >= 0) & (gy < IMG_H) & (gx >= 0) & (gx < IMG_W);
            if (valid) {
                uint32_t ldsoff = s_base + (uint32_t)(row * SCOLS + cx * 4) * 4u;
                uint32_t goff   = (uint32_t)((gy * IMG_W + gx) * 4);
                // GVS mode: SGPR64 base + per-lane 32-bit byte offset
                asm volatile("global_load_async_to_lds_b128 %0, %1, %2"
                             :: "v"(ldsoff), "v"(goff), "s"(src)
                             : "memory");
            }
        }
        asm volatile("s_wait_asynccnt 0" ::: "memory");
        __syncthreads();
    } else {
        // ---- interior tile: single TDM DMA of the whole 70x72 halo tile ----
        if (tid == 0) {
            const uint64_t gaddr = (uint64_t)(uintptr_t)src +
                (uint64_t)(uint32_t)(((y0 - HALO) * IMG_W + (x0 - 4)) * 4);

            // D# group 0: count=1 | lds_addr | global_addr(57b) | type=2
            u32x4 g0;
            g0[0] = 1u;                                    // count=1, user mode
            g0[1] = s_base;                                // lds_addr (bytes)
            g0[2] = (uint32_t)gaddr;                       // global_addr[31:0]
            g0[3] = (uint32_t)(gaddr >> 32) | 0x80000000u; // [56:32] | type=2

            // D# group 1 (bit-packed per ISA 8.4):
            i32x8 g1;
            g1[0] = 0x20000;                 // data_size=2 (4 bytes), no pad/iter
            g1[1] = (int)(IMG_W << 16);      // tensor_dim0[15:0] in [63:48]
            g1[2] = (int)(IMG_H << 16);      // dim0[31:16]=0, tensor_dim1[15:0]
            g1[3] = (int)(SCOLS << 16);      // dim1[31:16]=0, tile_dim0=72
            g1[4] = SROWS;                   // tile_dim1=70, tile_dim2=0
            g1[5] = IMG_W;                   // tensor_dim0_stride=512
            g1[6] = 0;                       // stride0[47:32]=0, stride1[15:0]=0
            g1[7] = (int)((IMG_W * IMG_H) >> 16); // tensor_dim1_stride[47:16]

            const i32x4 gz4 = {0, 0, 0, 0};            // groups 2/3 unused (2D)
            const i32x8 gz8 = {0, 0, 0, 0, 0, 0, 0, 0};
            // clang-23 / therock-10.0 signature: 6 args
            __builtin_amdgcn_tensor_load_to_lds(g0, g1, gz4, gz4, gz8, 0);
            __builtin_amdgcn_s_wait_tensorcnt(0);
        }
        __syncthreads();
    }

    // ---- horizontal 7-tap pass: T[y][x] = sum_k w[k] * S[y][x+1+k] ----
    for (int i = tid; i < SROWS * TILE; i += 256) {
        const int y  = i >> 6;
        const int xx = i & (TILE - 1);
        const float* r = &S[y * SCOLS + xx + 1];
        float acc = 0.0f;
        #pragma unroll
        for (int k = 0; k < KW; ++k) acc = __builtin_fmaf(w[k], r[k], acc);
        T[i] = acc;
    }
    __syncthreads();

    // ---- vertical 7-tap pass with register sliding window ----
    {
        const int xx   = tid & (TILE - 1);
        const int base = (tid >> 6) * 16;
        float win[KW];
        #pragma unroll
        for (int k = 0; k < KW; ++k) win[k] = T[(base + k) * TILE + xx];

        #pragma unroll 4
        for (int r = 0; r < 16; ++r) {
            float acc = 0.0f;
            #pragma unroll
            for (int k = 0; k < KW; ++k) acc = __builtin_fmaf(w[k], win[k], acc);
            // write-once output: NT store keeps L2 free for input halo reuse
            __builtin_nontemporal_store(
                acc, &dst[(size_t)(y0 + base + r) * IMG_W + (x0 + xx)]);
            if (r < 15) {
                #pragma unroll
                for (int k = 0; k < KW - 1; ++k) win[k] = win[k + 1];
                win[KW - 1] = T[(base + r + KW) * TILE + xx];
            }
        }
    }
}

extern "C" void kernel_launch(void* const* d_in, const int* in_sizes, int n_in,
                              void* d_out, int out_size, void* d_ws, size_t ws_size,
                              hipStream_t stream) {
    (void)n_in; (void)out_size; (void)d_ws; (void)ws_size;
    const float* x   = (const float*)d_in[0];
    const float* sig = (const float*)d_in[1];
    float*       out = (float*)d_out;

    const int planes = in_sizes[0] / (IMG_W * IMG_H);   // B*C = 96
    dim3 grid(IMG_W / TILE, IMG_H / TILE, planes);
    gauss7_tile_kernel<<<grid, 256, 0, stream>>>(x, sig, out);
}